// ElementwiseTensorProduct_63634235457618
// MI455X (gfx1250) — compile-verified
//
#include <hip/hip_runtime.h>
#include <math.h>

// ---------------------------------------------------------------------------
// Elementwise CG tensor product as a skinny WMMA GEMM:
//   rows   = N*128 channel-instances
//   A row  = p[36]  (pair products, padded 35->36)
//   B      = W^T [48][36] constant CG matrix (padded 35->48 cols)
//   D      = out[35], staged through LDS into e3nn concatenated layout,
//            then stored with fully coalesced 128B rounds.
// FP32 path via V_WMMA_F32_16X16X4_F32 (9 K-steps of 4, 3 N-tiles of 16).
// ---------------------------------------------------------------------------

typedef float v2f __attribute__((ext_vector_type(2)));
typedef float v8f __attribute__((ext_vector_type(8)));

#if defined(__has_builtin)
#if __has_builtin(__builtin_amdgcn_wmma_f32_16x16x4_f32)
#define HAVE_WMMA_F32X4 1
#endif
#endif

static __device__ inline v8f wmma_f32_16x16x4(v2f a, v2f b, v8f c) {
#ifdef HAVE_WMMA_F32X4
  return __builtin_amdgcn_wmma_f32_16x16x4_f32(
      /*neg_a=*/false, a, /*neg_b=*/false, b,
      /*c_mod=*/(short)0, c, /*reuse_a=*/false, /*reuse_b=*/false);
#else
  c[0] += a[0] * b[0] + a[1] * b[1];
  return c;
#endif
}

// ---------------------------------------------------------------------------
// Device-side generation of the CG matrix W^T (exact port of the reference's
// Racah formula + real<->complex spherical-harmonic change of basis).
// Single thread, double precision, ~microseconds, deterministic.
// ---------------------------------------------------------------------------

static __device__ double dfact(int n) {
  double r = 1.0;
  for (int i = 2; i <= n; ++i) r *= (double)i;
  return r;
}

static __device__ double su2_cg(int j1, int m1, int j2, int m2, int j3, int m3) {
  if (m1 + m2 != m3) return 0.0;
  int vmin = 0;
  if (-j1 + j2 + m3 > vmin) vmin = -j1 + j2 + m3;
  if (-j1 + m1 > vmin) vmin = -j1 + m1;
  int vmax = j2 + j3 + m1;
  if (j3 - j1 + j2 < vmax) vmax = j3 - j1 + j2;
  if (j3 + m3 < vmax) vmax = j3 + m3;
  double c = sqrt((2.0 * j3 + 1.0) *
                  (dfact(j3 + j1 - j2) * dfact(j3 - j1 + j2) * dfact(j1 + j2 - j3) *
                   dfact(j3 + m3) * dfact(j3 - m3)) /
                  (dfact(j1 + j2 + j3 + 1) * dfact(j1 - m1) * dfact(j1 + m1) *
                   dfact(j2 - m2) * dfact(j2 + m2)));
  double s = 0.0;
  for (int v = vmin; v <= vmax; ++v) {
    double t = (dfact(j2 + j3 + m1 - v) * dfact(j1 - m1 + v)) /
               (dfact(v) * dfact(j3 - j1 + j2 - v) * dfact(j3 + m3 - v) *
                dfact(v + j1 - j2 - m3));
    s += ((v + j2 + m2) & 1) ? -t : t;
  }
  return c * s;
}

static __device__ void build_cb(int l, double* qre, double* qim, int stride) {
  const int d = 2 * l + 1;
  for (int i = 0; i < d; ++i)
    for (int j = 0; j < d; ++j) { qre[i * stride + j] = 0.0; qim[i * stride + j] = 0.0; }
  const double r2 = 0.70710678118654752440;
  for (int m = -l; m < 0; ++m) {
    qre[(l + m) * stride + (l - m)] = r2;
    qim[(l + m) * stride + (l + m)] = -r2;
  }
  qre[l * stride + l] = 1.0;
  for (int m = 1; m <= l; ++m) {
    double sgn = (m & 1) ? -1.0 : 1.0;
    qre[(l + m) * stride + (l + m)] = sgn * r2;
    qim[(l + m) * stride + (l - m)] = sgn * r2;
  }
  double pre, pim;  // (-i)^l
  switch (l & 3) {
    case 0: pre = 1.0; pim = 0.0; break;
    case 1: pre = 0.0; pim = -1.0; break;
    case 2: pre = -1.0; pim = 0.0; break;
    default: pre = 0.0; pim = 1.0; break;
  }
  for (int i = 0; i < d; ++i)
    for (int j = 0; j < d; ++j) {
      double re = qre[i * stride + j], im = qim[i * stride + j];
      qre[i * stride + j] = re * pre - im * pim;
      qim[i * stride + j] = re * pim + im * pre;
    }
}

static __device__ void compute_block(int lo, int l, int pbase, int cbase, float* Wt) {
  const int d1 = 2 * lo + 1, d2 = 2 * l + 1;
  double C[9][5][5];
  for (int i = 0; i < d1; ++i)
    for (int k = 0; k < d2; ++k)
      for (int n = 0; n < d2; ++n) C[i][k][n] = 0.0;
  for (int m1 = -lo; m1 <= lo; ++m1)
    for (int m2 = -l; m2 <= l; ++m2) {
      int m3 = m1 + m2;
      if (m3 >= -l && m3 <= l)
        C[lo + m1][l + m2][l + m3] = su2_cg(lo, m1, l, m2, l, m3);
    }
  double Q1re[9][9], Q1im[9][9], Q2re[9][9], Q2im[9][9];
  build_cb(lo, &Q1re[0][0], &Q1im[0][0], 9);
  build_cb(l, &Q2re[0][0], &Q2im[0][0], 9);

  double O[9][5][5];
  double nrm = 0.0;
  for (int j = 0; j < d1; ++j)
    for (int li = 0; li < d2; ++li)
      for (int m = 0; m < d2; ++m) {
        double sre = 0.0;
        for (int i = 0; i < d1; ++i) {
          double are = Q1re[i][j], aim = Q1im[i][j];
          if (are == 0.0 && aim == 0.0) continue;
          for (int k = 0; k < d2; ++k) {
            double bre = are * Q2re[k][li] - aim * Q2im[k][li];
            double bim = are * Q2im[k][li] + aim * Q2re[k][li];
            if (bre == 0.0 && bim == 0.0) continue;
            for (int n = 0; n < d2; ++n) {
              double c = C[i][k][n];
              if (c == 0.0) continue;
              sre += c * (bre * Q2re[n][m] + bim * Q2im[n][m]);  // * conj(Q3)
            }
          }
        }
        O[j][li][m] = sre;
        nrm += sre * sre;
      }
  double scale = sqrt((double)(2 * lo + 1)) / sqrt(nrm);
  for (int j = 0; j < d1; ++j)
    for (int li = 0; li < d2; ++li)
      for (int m = 0; m < d2; ++m)
        Wt[(cbase + j) * 36 + (pbase + li * d2 + m)] =
            (float)(O[j][li][m] * scale);
}

__global__ void init_W_kernel(float* __restrict__ Wt) {
  if (threadIdx.x != 0 || blockIdx.x != 0) return;
  for (int i = 0; i < 48 * 36; ++i) Wt[i] = 0.0f;
  int cb = 0;
  compute_block(0, 0, /*pbase=*/0, cb, Wt); cb += 1;
  for (int lo = 0; lo <= 2; ++lo) { compute_block(lo, 1, 1, cb, Wt);  cb += 2 * lo + 1; }
  for (int lo = 0; lo <= 4; ++lo) { compute_block(lo, 2, 10, cb, Wt); cb += 2 * lo + 1; }
}

// ---------------------------------------------------------------------------
// Main WMMA kernel
// ---------------------------------------------------------------------------

// Output column c -> LDS staging cumulative offset (32-channel blocks),
// block stride (2lo+1), and k within block.
__device__ const int COL_CUM32[48] = {
    0, 32, 64, 64, 64, 160, 160, 160, 160, 160,
    320, 352, 352, 352, 448, 448, 448, 448, 448,
    608, 608, 608, 608, 608, 608, 608,
    832, 832, 832, 832, 832, 832, 832, 832, 832,
    0, 0, 0, 0, 0, 0, 0, 0, 0, 0, 0, 0, 0};
__device__ const int COL_STRIDE[48] = {
    1, 1, 3, 3, 3, 5, 5, 5, 5, 5,
    1, 3, 3, 3, 5, 5, 5, 5, 5,
    7, 7, 7, 7, 7, 7, 7,
    9, 9, 9, 9, 9, 9, 9, 9, 9,
    1, 1, 1, 1, 1, 1, 1, 1, 1, 1, 1, 1, 1};
__device__ const int COL_K[48] = {
    0, 0, 0, 1, 2, 0, 1, 2, 3, 4,
    0, 0, 1, 2, 0, 1, 2, 3, 4,
    0, 1, 2, 3, 4, 5, 6,
    0, 1, 2, 3, 4, 5, 6, 7, 8,
    0, 0, 0, 0, 0, 0, 0, 0, 0, 0, 0, 0, 0};

__global__ __launch_bounds__(256) void etp_wmma_kernel(
    const float* __restrict__ f1g, const float* __restrict__ f2g,
    const float* __restrict__ Wt, float* __restrict__ out) {
  // 8 waves/block; waves 0-3 -> sample 2*bid, waves 4-7 -> sample 2*bid+1.
  // Each wave owns 32 channels. LDS slice (1152 floats/wave) is used first
  // for A rows (32 x 36), then reused for output staging (1120 floats).
  __shared__ float buf[8 * 1152];
  const int tid = threadIdx.x;
  const int wave = tid >> 5;
  const int lane = tid & 31;
  const int n = blockIdx.x * 2 + (wave >> 2);
  const int ubase = (wave & 3) * 32;
  const int u = ubase + lane;
  float* ldsw = buf + wave * 1152;

  const float* f1 = f1g + (size_t)n * 1152;
  const float* f2 = f2g + (size_t)n * 1152;

  // ---- B fragments: whole 36x48 constant matrix per wave (54 VGPRs) ------
  const int kbh = (lane >> 4) << 1;  // 0 for lanes 0-15, 2 for lanes 16-31
  const int nlo = lane & 15;
  v2f bfrag[9][3];
#pragma unroll
  for (int t = 0; t < 9; ++t)
#pragma unroll
    for (int nt = 0; nt < 3; ++nt) {
      const float* bp = Wt + (size_t)(nt * 16 + nlo) * 36 + 4 * t + kbh;
      bfrag[t][nt] = *reinterpret_cast<const v2f*>(bp);
    }

  // ---- Phase 1: pair-product row p[36] -> LDS ----------------------------
  {
    float p[36];
    const float x0 = f1[u], y0 = f2[u];
    float x1[3], y1[3], x2[5], y2[5];
#pragma unroll
    for (int t = 0; t < 3; ++t) { x1[t] = f1[128 + 3 * u + t]; y1[t] = f2[128 + 3 * u + t]; }
#pragma unroll
    for (int t = 0; t < 5; ++t) { x2[t] = f1[512 + 5 * u + t]; y2[t] = f2[512 + 5 * u + t]; }
    p[0] = x0 * y0;
#pragma unroll
    for (int i = 0; i < 3; ++i)
#pragma unroll
      for (int j = 0; j < 3; ++j) p[1 + 3 * i + j] = x1[i] * y1[j];
#pragma unroll
    for (int i = 0; i < 5; ++i)
#pragma unroll
      for (int j = 0; j < 5; ++j) p[10 + 5 * i + j] = x2[i] * y2[j];
    p[35] = 0.0f;
#pragma unroll
    for (int q = 0; q < 9; ++q) {
      float4 v; v.x = p[4 * q]; v.y = p[4 * q + 1]; v.z = p[4 * q + 2]; v.w = p[4 * q + 3];
      *reinterpret_cast<float4*>(&ldsw[lane * 36 + 4 * q]) = v;
    }
  }
  __syncthreads();

  // ---- Phase 2: WMMA, both 16-row tiles kept in registers ----------------
  v8f acc[2][3];
#pragma unroll
  for (int rt = 0; rt < 2; ++rt)
#pragma unroll
    for (int nt = 0; nt < 3; ++nt)
      acc[rt][nt] = (v8f){0.f, 0.f, 0.f, 0.f, 0.f, 0.f, 0.f, 0.f};
#pragma unroll
  for (int rt = 0; rt < 2; ++rt) {
    const int rowA = nlo + 16 * rt;
#pragma unroll
    for (int t = 0; t < 9; ++t) {
      // A layout (16x4 f32): VGPR0 = K (lanes<16) / K+2 (lanes>=16), VGPR1 = +1
      v2f a = *reinterpret_cast<const v2f*>(&ldsw[rowA * 36 + 4 * t + kbh]);
#pragma unroll
      for (int nt = 0; nt < 3; ++nt)
        acc[rt][nt] = wmma_f32_16x16x4(a, bfrag[t][nt], acc[rt][nt]);
    }
  }
  __syncthreads();

  // ---- Phase 3: scatter D into LDS in output-memory order ----------------
  {
    int c_cum[3], c_st[3], c_k[3];
    bool c_ok[3];
#pragma unroll
    for (int nt = 0; nt < 3; ++nt) {
      const int c = nt * 16 + nlo;
      c_ok[nt] = (c < 35);
      c_cum[nt] = COL_CUM32[c];
      c_st[nt] = COL_STRIDE[c];
      c_k[nt] = COL_K[c];
    }
    const int mb = (lane >> 4) * 8;  // lanes 0-15 hold M=g, 16-31 hold M=g+8
#pragma unroll
    for (int rt = 0; rt < 2; ++rt)
#pragma unroll
      for (int nt = 0; nt < 3; ++nt) {
        if (!c_ok[nt]) continue;
#pragma unroll
        for (int g = 0; g < 8; ++g) {
          const int ul = 16 * rt + mb + g;  // local channel 0..31
          ldsw[c_cum[nt] + ul * c_st[nt] + c_k[nt]] = acc[rt][nt][g];
        }
      }
  }
  __syncthreads();

  // ---- Phase 4: coalesced stores: 9 contiguous runs, 35 x 128B rounds ----
  {
    // per-block: sample-offset, stride (2lo+1), LDS cumulative offset
    constexpr int NB = 9;
    constexpr int B_OFF[NB] = {0, 128, 256, 640, 1280, 1408, 1792, 2432, 3328};
    constexpr int B_S[NB]   = {1, 1, 3, 5, 1, 3, 5, 7, 9};
    constexpr int B_CUM[NB] = {0, 32, 64, 160, 320, 352, 448, 608, 832};
    float* outn = out + (size_t)n * 4480;
#pragma unroll
    for (int b = 0; b < NB; ++b) {
      float* outb = outn + B_OFF[b] + ubase * B_S[b];  // uniform base
#pragma unroll
      for (int r = 0; r < B_S[b]; ++r) {
        outb[(unsigned)(r * 32 + lane)] = ldsw[B_CUM[b] + r * 32 + lane];
      }
    }
  }
}

extern "C" void kernel_launch(void* const* d_in, const int* in_sizes, int n_in,
                              void* d_out, int out_size, void* d_ws, size_t ws_size,
                              hipStream_t stream) {
  const float* f1 = (const float*)d_in[0];
  const float* f2 = (const float*)d_in[1];
  float* out = (float*)d_out;
  float* Wt = (float*)d_ws;  // 48*36 floats

  const int nN = in_sizes[0] / 1152;  // 16384

  init_W_kernel<<<1, 32, 0, stream>>>(Wt);
  etp_wmma_kernel<<<nN / 2, 256, 0, stream>>>(f1, f2, Wt, out);
}